// Tree_53824530153725
// MI455X (gfx1250) — compile-verified
//
#include <hip/hip_runtime.h>

#define DEPTH  10
#define NLEAF  1024   // 2^DEPTH
#define FEAT   1024
#define BATCH  32768
#define OUTW   2048   // 2 + sum_{L=0..9} 2^{L+1} = 2*NLEAF

#define ROWS_PER_BLOCK 4          // one row per wave32
#define TPB (ROWS_PER_BLOCK * 32)

typedef int   v4i __attribute__((vector_size(16)));
typedef float v4f __attribute__((vector_size(16)));
typedef float v2f __attribute__((vector_size(8)));

typedef float e2f __attribute__((ext_vector_type(2)));   // WMMA f32 A/B (16x4 / 4x16)
typedef float e8f __attribute__((ext_vector_type(8)));   // WMMA f32 C/D (16x16)

#if defined(__gfx1250__) && __has_builtin(__builtin_amdgcn_global_load_async_to_lds_b128)
#define HAVE_ASYNC_LDS 1
#else
#define HAVE_ASYNC_LDS 0
#endif

static __device__ __forceinline__ void wait_async0() {
#if __has_builtin(__builtin_amdgcn_s_wait_asynccnt)
  __builtin_amdgcn_s_wait_asynccnt(0);
#else
  asm volatile("s_wait_asynccnt 0" ::: "memory");
#endif
}

static __device__ __forceinline__ float sigmoidf_(float t) {
  // 1/(1+e^-t) with v_rcp_f32 (1 ulp) instead of the IEEE div sequence
  return __builtin_amdgcn_rcpf(1.0f + __expf(-t));
}

// ---------------------------------------------------------------------------
// Phase 1 (WMMA): using_idx = iota @ feature_mask, exact in fp32.
//   idx[leaf] = sum_f f * mask[f][leaf]   (mask columns are one-hot)
// One wave per 16-leaf tile (64 waves). Per K-step of 4:
//   A (16x4): every row = {f0, f0+1, f0+2, f0+3}   (broadcast iota)
//   B (4x16): mask[f0+k][leaf0 + n]
// Accumulate 256 steps of v_wmma_f32_16x16x4_f32; row 0 of D = the 16 indices.
// A-slot/B-slot K labels use the mirrored 32-bit layout (VGPR i, half h -> k =
// 2h + i); the iota trick is invariant to any consistent K-slot permutation.
// ---------------------------------------------------------------------------
__global__ __launch_bounds__(256) void extract_idx_wmma(
    const float* __restrict__ mask, int* __restrict__ idx) {
  const int wave  = (blockIdx.x * blockDim.x + threadIdx.x) >> 5;  // 0..63
  const int lane  = threadIdx.x & 31;
  const int leaf0 = wave << 4;          // 16 leaves per wave
  const int half  = lane >> 4;          // lanes 16..31 hold K base+2
  const int n     = lane & 15;          // leaf within tile

  e8f acc = {};                         // C = 0
  const float kbase = (float)(2 * half);
  #pragma unroll 4
  for (int f0 = 0; f0 < FEAT; f0 += 4) {
    e2f a;
    a.x = (float)f0 + kbase;            // k = 2h
    a.y = (float)f0 + kbase + 1.0f;     // k = 2h+1
    const float* bp = mask + (size_t)(f0 + 2 * half) * NLEAF + leaf0 + n;
    e2f b;
    b.x = bp[0];                        // row f0+2h
    b.y = bp[NLEAF];                    // row f0+2h+1
    acc = __builtin_amdgcn_wmma_f32_16x16x4_f32(
        false, a, false, b, (short)0, acc, false, false);
  }
  // D VGPR0: lanes 0..15 hold row M=0, N=lane  -> the 16 leaf indices
  if (lane < 16) idx[leaf0 + lane] = (int)(acc[0] + 0.5f);
}

// ---------------------------------------------------------------------------
// Phase 2: gather + sigmoid + heap product. One row per wave32.
// out[c] = out[c>>1] * (c&1 ? 1-d[c>>1] : d[c>>1]),  d[n] = sigmoid(x[idx[n]])
// Layer L (L>=1) output columns [2^{L+1}, 2^{L+2}); lane owns pair m:
// reads mu[m], writes mu[2m],mu[2m+1] in place (descending chunks = safe).
// ---------------------------------------------------------------------------
__global__ __launch_bounds__(TPB) void tree_kernel(const float* __restrict__ x,
                                                   const int* __restrict__ idx,
                                                   float* __restrict__ out) {
  __shared__ int   s_idx[NLEAF];                  // 4 KB, block-shared
  __shared__ float s_x [ROWS_PER_BLOCK][FEAT];    // 16 KB, per-wave row
  __shared__ float s_mu[ROWS_PER_BLOCK][NLEAF];   // 16 KB, per-wave tree

  const int tid  = threadIdx.x;
  const int w    = tid >> 5;
  const int lane = tid & 31;
  const int b    = blockIdx.x * ROWS_PER_BLOCK + w;

  // block-wide: stage idx into LDS
  for (int i = tid; i < NLEAF; i += TPB) s_idx[i] = idx[i];

  // per-wave: stage this row of x (4 KB) into LDS, coalesced b128 per lane
  const float* gx = x + (size_t)b * FEAT;
  float* lx = s_x[w];
#if HAVE_ASYNC_LDS
  #pragma unroll
  for (int it = 0; it < 8; ++it) {
    int off = (it * 32 + lane) * 4;               // float index, 16B aligned
    __builtin_amdgcn_global_load_async_to_lds_b128(
        (v4i*)(gx + off), (v4i*)(lx + off), 0, 0);
  }
  wait_async0();
#else
  #pragma unroll
  for (int it = 0; it < 8; ++it) {
    int off = it * 32 + lane;
    reinterpret_cast<v4f*>(lx)[off] =
        reinterpret_cast<const v4f*>(gx)[off];
  }
#endif
  __syncthreads();   // s_idx is cross-wave; x/mu are wave-private

  float* mu   = s_mu[w];
  float* orow = out + (size_t)b * OUTW;

  // root: columns 0..3 = {1, 1, d1, 1-d1}; seed mu[0..1]
  if (lane == 0) {
    float d1 = sigmoidf_(lx[s_idx[1]]);
    mu[0] = d1;
    mu[1] = 1.0f - d1;
    v4f h = {1.0f, 1.0f, d1, 1.0f - d1};
    __builtin_nontemporal_store(h, reinterpret_cast<v4f*>(orow));
  }
  // per-wave in-order LDS: lane0's stores complete before later loads

  #pragma unroll
  for (int L = 1; L < DEPTH; ++L) {
    const int pairs = 1 << L;                 // nodes this layer
    const int nch   = (pairs + 31) >> 5;
    for (int t = nch - 1; t >= 0; --t) {      // descending => in-place safe
      const int m = (t << 5) + lane;
      if (m < pairs) {
        const float parent = mu[m];
        const float xv     = lx[s_idx[pairs + m]];   // node column = 2^L + m
        const float dv     = sigmoidf_(xv);
        const float e      = parent * dv;
        const float o      = parent * (1.0f - dv);
        mu[2 * m]     = e;
        mu[2 * m + 1] = o;
        v2f st = {e, o};
        __builtin_nontemporal_store(
            st, reinterpret_cast<v2f*>(orow + 2 * pairs + 2 * m));
      }
    }
  }
}

// ---------------------------------------------------------------------------
extern "C" void kernel_launch(void* const* d_in, const int* in_sizes, int n_in,
                              void* d_out, int out_size, void* d_ws, size_t ws_size,
                              hipStream_t stream) {
  const float* x    = (const float*)d_in[0];   // (BATCH, FEAT)
  const float* mask = (const float*)d_in[1];   // (FEAT, NLEAF)
  float*       out  = (float*)d_out;           // (BATCH, OUTW)
  int*         idx  = (int*)d_ws;              // NLEAF ints of scratch

  extract_idx_wmma<<<dim3(8), dim3(256), 0, stream>>>(mask, idx);  // 64 waves
  tree_kernel<<<dim3(BATCH / ROWS_PER_BLOCK), dim3(TPB), 0, stream>>>(x, idx, out);
}